// NousNumberTheoryBranch_85126251807083
// MI455X (gfx1250) — compile-verified
//
#include <hip/hip_runtime.h>

// Clang ext-vector types so __builtin_nontemporal_{load,store} accept them
// (HIP's int4/float4 are class types, which those builtins reject).
typedef int   v4i __attribute__((ext_vector_type(4)));
typedef float v4f __attribute__((ext_vector_type(4)));

// ---------------------------------------------------------------------------
// Binary (Stein) GCD: no integer division. Inputs here are < 2^20, so the
// loop runs at most ~40 subtract/shift steps; each step is 5 VALU ops
// (v_ctz, v_lshrrev, v_min_u32, v_max_u32, v_sub). Early-finished lanes drop
// out via EXEC masking. ctz is only ever applied to nonzero values (no UB).
// ---------------------------------------------------------------------------
__device__ __forceinline__ unsigned dev_gcd(unsigned u, unsigned v) {
    if (u == 0u) return v;
    if (v == 0u) return u;
    const unsigned shift = (unsigned)__builtin_ctz(u | v);
    u >>= __builtin_ctz(u);                 // u odd
    do {
        v >>= __builtin_ctz(v);             // v odd
        const unsigned t = (u < v) ? u : v; // v_min_u32
        const unsigned m = (u > v) ? u : v; // v_max_u32
        v = m - t;                          // even (or 0)
        u = t;                              // odd
    } while (v != 0u);
    return u << shift;
}

// ---------------------------------------------------------------------------
// Exact a % p for 0 <= a < 2^24, 1 <= p < 2^24 via v_rcp_f32.
// a and p are exactly representable in f32; rel. error of a*rcp(p) is
// <= ~2^-22, so |q_est - a/p| < 0.25 and trunc(q_est) is within +/-1 of the
// true quotient. Two conditional fixups make the remainder exact.
// ---------------------------------------------------------------------------
__device__ __forceinline__ int dev_mod(int a, int p) {
    const float q = (float)a * __builtin_amdgcn_rcpf((float)p);
    int r = a - (int)q * p;
    r = (r < 0)  ? r + p : r;
    r = (r >= p) ? r - p : r;
    return r;
}

// Main vectorized kernel: 4 elements per lane, 128-bit NT loads/stores.
__global__ __launch_bounds__(256) void nt_gcdmod_vec4(
    const v4i* __restrict__ a4, const v4i* __restrict__ b4,
    const v4i* __restrict__ p4,
    const float* __restrict__ Wg, const float* __restrict__ Wm,
    v4f* __restrict__ outg4, v4f* __restrict__ outm4, int n4)
{
    const int i = blockIdx.x * 256 + threadIdx.x;
    if (i >= n4) return;

    const float wg = Wg[0];   // uniform -> s_load_b32
    const float wm = Wm[0];

    const v4i av = __builtin_nontemporal_load(a4 + i);  // global_load_b128 NT
    const v4i bv = __builtin_nontemporal_load(b4 + i);
    const v4i pv = __builtin_nontemporal_load(p4 + i);

    v4f g, m;
    g.x = (float)dev_gcd((unsigned)av.x, (unsigned)bv.x) * wg;
    g.y = (float)dev_gcd((unsigned)av.y, (unsigned)bv.y) * wg;
    g.z = (float)dev_gcd((unsigned)av.z, (unsigned)bv.z) * wg;
    g.w = (float)dev_gcd((unsigned)av.w, (unsigned)bv.w) * wg;

    m.x = (float)dev_mod(av.x, pv.x) * wm;
    m.y = (float)dev_mod(av.y, pv.y) * wm;
    m.z = (float)dev_mod(av.z, pv.z) * wm;
    m.w = (float)dev_mod(av.w, pv.w) * wm;

    __builtin_nontemporal_store(g, outg4 + i);          // global_store_b128 NT
    __builtin_nontemporal_store(m, outm4 + i);
}

// Scalar tail for n % 4 != 0 (not hit for N = 2^25, kept for robustness).
__global__ void nt_gcdmod_tail(
    const int* __restrict__ a, const int* __restrict__ b,
    const int* __restrict__ p,
    const float* __restrict__ Wg, const float* __restrict__ Wm,
    float* __restrict__ outg, float* __restrict__ outm, int base, int n)
{
    const int i = base + blockIdx.x * blockDim.x + threadIdx.x;
    if (i >= n) return;
    const float wg = Wg[0];
    const float wm = Wm[0];
    outg[i] = (float)dev_gcd((unsigned)a[i], (unsigned)b[i]) * wg;
    outm[i] = (float)dev_mod(a[i], p[i]) * wm;
}

extern "C" void kernel_launch(void* const* d_in, const int* in_sizes, int n_in,
                              void* d_out, int out_size, void* d_ws, size_t ws_size,
                              hipStream_t stream) {
    (void)n_in; (void)out_size; (void)d_ws; (void)ws_size;

    const int*   a  = (const int*)d_in[0];
    const int*   b  = (const int*)d_in[1];
    const int*   p  = (const int*)d_in[2];
    const float* Wg = (const float*)d_in[3];
    const float* Wm = (const float*)d_in[4];

    const int n = in_sizes[0];
    float* outg = (float*)d_out;       // gcd branch: d_out[0 .. n)
    float* outm = (float*)d_out + n;   // mod branch: d_out[n .. 2n)

    const int n4  = n >> 2;
    const int rem = n & 3;

    if (n4 > 0) {
        const int blocks = (n4 + 255) / 256;
        nt_gcdmod_vec4<<<blocks, 256, 0, stream>>>(
            (const v4i*)a, (const v4i*)b, (const v4i*)p, Wg, Wm,
            (v4f*)outg, (v4f*)outm, n4);
    }
    if (rem > 0) {
        nt_gcdmod_tail<<<1, 32, 0, stream>>>(a, b, p, Wg, Wm, outg, outm,
                                             n4 << 2, n);
    }
}